// GCN_5901285065199
// MI455X (gfx1250) — compile-verified
//
#include <hip/hip_runtime.h>

typedef __attribute__((ext_vector_type(2))) float v2f;
typedef __attribute__((ext_vector_type(4))) float v4f;
typedef __attribute__((ext_vector_type(8))) float v8f;

#define N_NODES 100000
#define N_EDGES 1600000
#define D 128
#define STRIPS (N_NODES / 16)   // 6250 16-row strips, exact

// ---------------------------------------------------------------------------
// Zero-fill (vectorized b128 stores)
// ---------------------------------------------------------------------------
__global__ __launch_bounds__(256) void gcn_zero(float* __restrict__ p, int n4) {
    int i = blockIdx.x * blockDim.x + threadIdx.x;
    if (i < n4) ((v4f*)p)[i] = (v4f){0.f, 0.f, 0.f, 0.f};
}

// ---------------------------------------------------------------------------
// COO SpMM: out[rows[e]] += vals[e] * h[cols[e]]
// One wave32 per edge; each lane handles 4 contiguous features (float4 gather
// -> global_load_b128, then 4x global_atomic_add_f32). h (51.2 MB) lives in
// the 192 MB L2, so gathers and atomics resolve on-die. Edge metadata is
// wave-uniform: force it into SGPRs so addresses compute on the scalar pipe.
// ---------------------------------------------------------------------------
__global__ __launch_bounds__(256) void gcn_spmm(const float* __restrict__ h,
                                                const float* __restrict__ vals,
                                                const int*   __restrict__ rows,
                                                const int*   __restrict__ cols,
                                                float*       __restrict__ out) {
    int tid  = blockIdx.x * blockDim.x + threadIdx.x;
    int e    = tid >> 5;
    int lane = tid & 31;
    if (e >= N_EDGES) return;

    // e is uniform across the wave; broadcast loads into SGPRs.
    int r = __builtin_amdgcn_readfirstlane(rows[e]);
    int c = __builtin_amdgcn_readfirstlane(cols[e]);
    int vbits = __builtin_amdgcn_readfirstlane(__float_as_int(vals[e]));
    float v = __int_as_float(vbits);

    const float4* src = (const float4*)(h + (size_t)c * D) + lane;
    float4 hv = *src;

    float* dst = out + (size_t)r * D + lane * 4;
    atomicAdd(dst + 0, v * hv.x);
    atomicAdd(dst + 1, v * hv.y);
    atomicAdd(dst + 2, v * hv.z);
    atomicAdd(dst + 3, v * hv.w);
}

// ---------------------------------------------------------------------------
// Dense GEMM: out = act(H[100000x128] @ W[128x128]) via V_WMMA_F32_16X16X4_F32
// Block = 256 threads = 8 waves. W staged in LDS *K-pair interleaved*:
//   ldsW2[(r*D + n)*2 + p] = W[2r + p][n]      (r = K/2, p = 0/1)
// so a B fragment {W[K][n], W[K+1][n]} is one contiguous, 8B-aligned pair ->
// single ds_load_b64 directly into an even VGPR pair (no register shuffles).
// Each wave computes a 16-row strip across all 8 column tiles: per K-step one
// global_load_b64 A-fragment reused across 8 WMMAs; 32 K-steps * 8 = 256 wmma.
// ---------------------------------------------------------------------------
__global__ __launch_bounds__(256) void gcn_gemm(const float* __restrict__ H,
                                                const float* __restrict__ W,
                                                float*       __restrict__ out,
                                                int relu) {
    __shared__ __align__(16) float ldsW2[D * D];   // 64 KB, pair-interleaved

    int t = threadIdx.x;
    {
        // Cooperative interleave: thread t handles row-pair r = t>>2 and a
        // 32-column chunk. 2 float4 loads -> 4 v2f (b64) interleaved stores.
        int r  = t >> 2;              // 0..63  (K = 2r, 2r+1)
        int c0 = (t & 3) * 32;        // column chunk base
        const float* w0 = W + (size_t)(2 * r) * D;
        const float* w1 = w0 + D;
        v2f* dst = (v2f*)ldsW2 + (size_t)r * D;
        #pragma unroll
        for (int n8 = 0; n8 < 8; ++n8) {
            int n = c0 + n8 * 4;
            v4f a0 = *(const v4f*)(w0 + n);
            v4f a1 = *(const v4f*)(w1 + n);
            dst[n + 0] = (v2f){a0.x, a1.x};
            dst[n + 1] = (v2f){a0.y, a1.y};
            dst[n + 2] = (v2f){a0.z, a1.z};
            dst[n + 3] = (v2f){a0.w, a1.w};
        }
    }
    __syncthreads();

    int wave  = t >> 5;
    int lane  = t & 31;
    int strip = blockIdx.x * 8 + wave;
    if (strip >= STRIPS) return;           // wave-uniform; EXEC stays all-1s

    int m0   = strip * 16;
    int half = lane >> 4;                  // 0: K=k,k+1 ; 1: K=k+2,k+3
    int l    = lane & 15;

    const float* arow = H + (size_t)(m0 + l) * D + 2 * half;
    const v2f*   bbase = (const v2f*)ldsW2 + (size_t)half * D + l;

    v8f acc[8];
    #pragma unroll
    for (int j = 0; j < 8; ++j) acc[j] = (v8f){0.f,0.f,0.f,0.f,0.f,0.f,0.f,0.f};

    for (int k = 0; k < D; k += 4) {
        // A fragment (16x4 f32): lanes 0-15 hold K=k,k+1; lanes 16-31 K=k+2,k+3
        v2f a = *(const v2f*)(arow + k);
        const v2f* brow = bbase + (size_t)(k >> 1) * D;   // row-pair k/2 (+half)
        #pragma unroll
        for (int j = 0; j < 8; ++j) {
            v2f b = brow[j * 16];          // {W[K][n], W[K+1][n]} in one b64
            acc[j] = __builtin_amdgcn_wmma_f32_16x16x4_f32(
                /*neg_a=*/false, a, /*neg_b=*/false, b,
                /*c_mod=*/(short)0, acc[j],
                /*reuse_a=*/false, /*reuse_b=*/false);
        }
    }

    // C layout: VGPR i holds row m0 + 8*half + i, col j*16 + l
    #pragma unroll
    for (int j = 0; j < 8; ++j) {
        float* o = out + (size_t)(m0 + 8 * half) * D + j * 16 + l;
        #pragma unroll
        for (int i = 0; i < 8; ++i) {
            float x = acc[j][i];
            if (relu) x = fmaxf(x, 0.f);
            o[(size_t)i * D] = x;
        }
    }
}

// ---------------------------------------------------------------------------
// Launch sequence (FACTOR == 1.0, so SpMM scaling is a no-op):
//   A = spmm(x); B = spmm(A); A = relu(B @ W1);
//   B = spmm(A); A = spmm(B); out = A @ W2
// ---------------------------------------------------------------------------
extern "C" void kernel_launch(void* const* d_in, const int* in_sizes, int n_in,
                              void* d_out, int out_size, void* d_ws, size_t ws_size,
                              hipStream_t stream) {
    const float* x    = (const float*)d_in[0];
    const float* vals = (const float*)d_in[1];
    const int*   rows = (const int*)d_in[2];
    const int*   cols = (const int*)d_in[3];
    const float* W1   = (const float*)d_in[4];
    const float* W2   = (const float*)d_in[5];
    float*       out  = (float*)d_out;

    float* A = (float*)d_ws;
    float* B = A + (size_t)N_NODES * D;

    const int n4    = N_NODES * D / 4;                      // float4 count
    const int zgrid = (n4 + 255) / 256;                     // 12500 blocks
    const int sgrid = (int)(((size_t)N_EDGES * 32) / 256);  // 200000 blocks
    const int ggrid = (STRIPS + 7) / 8;                     // 782 blocks

    // Layer 1: two propagations + relu(h @ W1)
    gcn_zero<<<zgrid, 256, 0, stream>>>(A, n4);
    gcn_spmm<<<sgrid, 256, 0, stream>>>(x, vals, rows, cols, A);
    gcn_zero<<<zgrid, 256, 0, stream>>>(B, n4);
    gcn_spmm<<<sgrid, 256, 0, stream>>>(A, vals, rows, cols, B);
    gcn_gemm<<<ggrid, 256, 0, stream>>>(B, W1, A, /*relu=*/1);

    // Layer 2: two propagations + h @ W2
    gcn_zero<<<zgrid, 256, 0, stream>>>(B, n4);
    gcn_spmm<<<sgrid, 256, 0, stream>>>(A, vals, rows, cols, B);
    gcn_zero<<<zgrid, 256, 0, stream>>>(A, n4);
    gcn_spmm<<<sgrid, 256, 0, stream>>>(B, vals, rows, cols, A);
    gcn_gemm<<<ggrid, 256, 0, stream>>>(A, W2, out, /*relu=*/0);
}